// RegressionHead_9139690406424
// MI455X (gfx1250) — compile-verified
//
#include <hip/hip_runtime.h>
#include <hip/hip_bf16.h>

#define W      256
#define WP     257          // W + 1 (dustbin)
#define STRIDE 258          // padded LDS row stride (floats): stride-2 banks
#define ITERS  10

typedef __attribute__((ext_vector_type(2))) float v2f;
typedef __attribute__((ext_vector_type(8))) float v8f;

// LDS byte offset of a __shared__ object (generic -> AS3 -> int)
__device__ __forceinline__ unsigned lds_addr_of(const void* p) {
  return (unsigned)(unsigned long long)(__attribute__((address_space(3))) const void*)p;
}

// CDNA5 async global->LDS copy, 16B per lane, tracked by ASYNCcnt.
__device__ __forceinline__ void async_load16(unsigned lds, const void* g) {
  asm volatile("global_load_async_to_lds_b128 %0, %1, off"
               :: "v"(lds), "v"((unsigned long long)g) : "memory");
}

__global__ __launch_bounds__(1024, 1)
void sinkhorn_disp_kernel(const float* __restrict__ attn,
                          const float* __restrict__ phi_p,
                          float* __restrict__ out, int nhw) {
  __shared__ float tile[WP][STRIDE];   // 265,224 B — needs CDNA5 320KB LDS
  __shared__ float u[WP];
  __shared__ float v[WP];
  __shared__ float rmax[WP];

  const int tid  = threadIdx.x;
  const int lane = tid & 31;
  const int wid  = tid >> 5;
  const int slice = blockIdx.x;
  const float* base = attn + (size_t)slice * W * W;
  const float phi = phi_p[0];

  const float LOG_2W   =  6.23832463f;   // log(512)
  const float NLOG_2W  = -6.23832463f;   // log_mu / log_nu for j < W
  const float LOG_HALF = -0.69314718f;   // log_mu / log_nu dustbin

  // ---- Phase 0: stage slice into LDS via async tensor path ----
  // 256 rows x 64 x 16B chunks = 16384 chunks, 16 per thread.
  #pragma unroll
  for (int k = 0; k < 16; ++k) {
    int chunk = tid + (k << 10);
    int r  = chunk >> 6;
    int c4 = (chunk & 63) << 2;                 // column (floats)
    async_load16(lds_addr_of(&tile[r][c4]), base + (size_t)r * W + c4);
  }
  if (tid < W)  tile[tid][W] = phi;             // dustbin column
  if (tid < WP) tile[W][tid] = phi;             // dustbin row
  if (tid < WP) { u[tid] = 0.f; v[tid] = 0.f; }
  __builtin_amdgcn_s_wait_asynccnt(0);
  __syncthreads();

  // ---- Phase 1: 10 Sinkhorn iterations, fully LDS-resident ----
  for (int it = 0; it < ITERS; ++it) {
    // v[j] = log_nu[j] - lse_i( tile[i][j] + u[i] ) : one wave per column,
    // lanes stride rows (stride-258 -> conflict-free banks).
    for (int j = wid; j < WP; j += 32) {
      float xv[9];
      float m = -3.0e38f;
      #pragma unroll
      for (int t = 0; t < 9; ++t) {
        int i = lane + (t << 5);
        int ic = (i < WP) ? i : (WP - 1);
        float x = tile[ic][j] + u[ic];
        xv[t] = (i < WP) ? x : -3.0e38f;
        m = fmaxf(m, xv[t]);
      }
      #pragma unroll
      for (int d = 16; d >= 1; d >>= 1) m = fmaxf(m, __shfl_xor(m, d, 32));
      float s = 0.f;
      #pragma unroll
      for (int t = 0; t < 9; ++t) s += __expf(xv[t] - m);
      #pragma unroll
      for (int d = 16; d >= 1; d >>= 1) s += __shfl_xor(s, d, 32);
      if (lane == 0) {
        float lnu = (j < W) ? NLOG_2W : LOG_HALF;
        v[j] = lnu - (m + __logf(s));
      }
    }
    __syncthreads();

    // u[i] = log_mu[i] - lse_j( tile[i][j] + v[j] ) : wave owns 16 rows,
    // WMMA f32 16x16x4 (A = exp values, B = ones) does the row sums on the
    // matrix pipe while VALU produces the exps.
    if (wid < 17) {
      const int r0  = wid << 4;
      const int row = r0 + (lane & 15);
      const int rc  = (row < WP) ? row : W;     // clamp for the partial wave
      const int kp  = lane >> 4;                // which K pair this lane feeds

      // pass 1: per-row max (lanes L and L+16 cover disjoint column sets)
      float m = -3.0e38f;
      for (int t = 0; t <= 64; ++t) {
        int c0 = (t << 2) + (kp << 1);
        int c1 = c0 + 1;
        if (c0 <= W) m = fmaxf(m, tile[rc][c0] + v[c0]);
        if (c1 <= W) m = fmaxf(m, tile[rc][c1] + v[c1]);
      }
      m = fmaxf(m, __shfl_xor(m, 16, 32));
      if (lane < 16 && row < WP) rmax[row] = m;

      // pass 2: sum of exp via WMMA accumulate
      v8f acc = {};
      v2f ones; ones.x = 1.f; ones.y = 1.f;
      for (int t = 0; t <= 64; ++t) {
        int c0 = (t << 2) + (kp << 1);
        int c1 = c0 + 1;
        int cc0 = (c0 <= W) ? c0 : W;
        int cc1 = (c1 <= W) ? c1 : W;
        float e0 = __expf(tile[rc][cc0] + v[cc0] - m);
        float e1 = __expf(tile[rc][cc1] + v[cc1] - m);
        v2f a; a.x = (c0 <= W) ? e0 : 0.f; a.y = (c1 <= W) ? e1 : 0.f;
        acc = __builtin_amdgcn_wmma_f32_16x16x4_f32(
            false, a, false, ones, (short)0, acc, false, false);
      }
      // D layout: VGPR q, lanes 0-15 -> row r0+q, lanes 16-31 -> row r0+8+q
      if ((lane & 15) == 0) {
        int rbase = r0 + ((lane >> 4) << 3);
        #pragma unroll
        for (int q = 0; q < 8; ++q) {
          int r = rbase + q;
          if (r < WP) {
            float lmu = (r < W) ? NLOG_2W : LOG_HALF;
            u[r] = lmu - (rmax[r] + __logf(acc[q]));
          }
        }
      }
    }
    __syncthreads();
  }

  // ---- Phase 2: argmax + 3-tap soft-argmax disparity, occlusion ----
  for (int i = wid; i < W; i += 32) {
    const float ui = u[i];
    float best = -3.0e38f; int bestj = 0;
    #pragma unroll
    for (int t = 0; t < 8; ++t) {
      int j = lane + (t << 5);
      float val = __expf(tile[i][j] + ui + v[j] + LOG_2W);
      if (val > best) { best = val; bestj = j; }      // keeps first max per lane
    }
    #pragma unroll
    for (int d = 16; d >= 1; d >>= 1) {               // first-occurrence tiebreak
      float ov = __shfl_xor(best, d, 32);
      int   oj = __shfl_xor(bestj, d, 32);
      if (ov > best || (ov == best && oj < bestj)) { best = ov; bestj = oj; }
    }
    if (lane == 0) {
      float a[3], p[3];
      #pragma unroll
      for (int dd = 0; dd < 3; ++dd) {
        int jj = bestj + dd - 1;
        bool ok = (jj >= 0) && (jj < W);
        int jc = ok ? jj : 0;
        float av = __expf(tile[i][jc] + ui + v[jc] + LOG_2W);
        a[dd] = ok ? av : 0.f;
        float pv = (float)(i - jj);
        p[dd] = (ok && pv > 0.f) ? pv : 0.f;
      }
      float norm = a[0] + a[1] + a[2];
      float nw = (norm < 0.1f) ? 1.0f : norm;
      float disp = (a[0]*p[0] + a[1]*p[1] + a[2]*p[2]) / nw;
      out[(size_t)slice * W + i]                    = disp;       // disp_pred
      out[(size_t)nhw + (size_t)slice * W + i]      = 1.0f - nw;  // occ_pred
    }
  }
}

extern "C" void kernel_launch(void* const* d_in, const int* in_sizes, int n_in,
                              void* d_out, int out_size, void* d_ws, size_t ws_size,
                              hipStream_t stream) {
  const float* attn = (const float*)d_in[0];
  const float* phi  = (const float*)d_in[1];
  float* out = (float*)d_out;
  int slices = in_sizes[0] / (W * W);   // N*H = 480
  int nhw = slices * W;
  sinkhorn_disp_kernel<<<slices, 1024, 0, stream>>>(attn, phi, out, nhw);
}